// MambaCross_84378927497377
// MI455X (gfx1250) — compile-verified
//
#include <hip/hip_runtime.h>
#include <math.h>

typedef float v2f __attribute__((ext_vector_type(2)));
typedef float v8f __attribute__((ext_vector_type(8)));

#define BATCHN 16
#define LSEQ   512
#define FDIM   256
#define DMODEL 512
#define NSTATE 16
#define KCONV  4
#define NLAYER 2
#define RKDIM  32
#define ROWS   (BATCHN * LSEQ)   // 8192
#define TCHUNK 8

// ---------------------------------------------------------------------------
// fp32 WMMA GEMM, register-blocked: one wave computes a 32x32 C tile as 2x2
// fragments of V_WMMA_F32_16X16X4_F32 (K step 4).  Per k-step: 2 A-frag + 2
// B-frag loads feed 4 WMMAs (1 load/WMMA).  Layout per CDNA5 ISA 7.12.2:
//   A 16x4 f32 : lanes 0-15 -> {K=0,K=1}, lanes 16-31 -> {K=2,K=3}, M=lane&15
//   B 4x16 f32 : mirrored (lanes 0-15 rows K=0/1, lanes 16-31 rows K=2/3)
//   C 16x16    : VGPR r -> rows r (lanes 0-15) and r+8 (lanes 16-31)
// TRANSB=1 means B is stored row-major N x K (multiply by B^T).
// ---------------------------------------------------------------------------
template <bool TRANSB, bool ACC>
__global__ __launch_bounds__(256) void gemm_wmma_f32(
    const float* __restrict__ A, const float* __restrict__ B,
    float* __restrict__ C, int K, int lda, int ldb, int ldc,
    long sA, long sB, long sC)
{
    const int lane = threadIdx.x;       // 0..31
    const int half = lane >> 4;
    const int lp   = lane & 15;
    const int n0   = blockIdx.x * 32;
    const int m0   = (blockIdx.y * 8 + threadIdx.y) * 32;
    const float* Ab = A + (long)blockIdx.z * sA;
    const float* Bb = B + (long)blockIdx.z * sB;
    float*       Cb = C + (long)blockIdx.z * sC;

    v8f acc00 = {0.f,0.f,0.f,0.f,0.f,0.f,0.f,0.f};
    v8f acc01 = acc00, acc10 = acc00, acc11 = acc00;
    const long ar0 = m0 + lp,      ar1 = m0 + 16 + lp;
    const long bc0 = n0 + lp,      bc1 = n0 + 16 + lp;

    for (int k0 = 0; k0 < K; k0 += 4) {
        const int kk = k0 + 2 * half;
        if ((k0 & 63) == 0 && k0 + 64 < K) {
            __builtin_prefetch(Ab + ar0 * lda + kk + 64, 0, 1);
            if (TRANSB) __builtin_prefetch(Bb + bc0 * ldb + kk + 64, 0, 1);
        }
        v2f a0, a1, b0, b1;
        { const float* p = Ab + ar0 * lda + kk; a0.x = p[0]; a0.y = p[1]; }
        { const float* p = Ab + ar1 * lda + kk; a1.x = p[0]; a1.y = p[1]; }
        if (TRANSB) {
            const float* p0 = Bb + bc0 * ldb + kk; b0.x = p0[0]; b0.y = p0[1];
            const float* p1 = Bb + bc1 * ldb + kk; b1.x = p1[0]; b1.y = p1[1];
        } else {
            b0.x = Bb[(long)kk * ldb + bc0]; b0.y = Bb[(long)(kk + 1) * ldb + bc0];
            b1.x = Bb[(long)kk * ldb + bc1]; b1.y = Bb[(long)(kk + 1) * ldb + bc1];
        }
        acc00 = __builtin_amdgcn_wmma_f32_16x16x4_f32(false, a0, false, b0, (short)0, acc00, false, false);
        acc01 = __builtin_amdgcn_wmma_f32_16x16x4_f32(false, a0, false, b1, (short)0, acc01, false, false);
        acc10 = __builtin_amdgcn_wmma_f32_16x16x4_f32(false, a1, false, b0, (short)0, acc10, false, false);
        acc11 = __builtin_amdgcn_wmma_f32_16x16x4_f32(false, a1, false, b1, (short)0, acc11, false, false);
    }
#pragma unroll
    for (int r = 0; r < 8; ++r) {
        const int row0 = m0 + r + 8 * half;
        const int row1 = row0 + 16;
        long i00 = (long)row0 * ldc + n0 + lp;
        long i01 = i00 + 16;
        long i10 = (long)row1 * ldc + n0 + lp;
        long i11 = i10 + 16;
        if (ACC) {
            Cb[i00] += acc00[r]; Cb[i01] += acc01[r];
            Cb[i10] += acc10[r]; Cb[i11] += acc11[r];
        } else {
            Cb[i00] = acc00[r]; Cb[i01] = acc01[r];
            Cb[i10] = acc10[r]; Cb[i11] = acc11[r];
        }
    }
}

// ---------------------------------------------------------------------------
__global__ __launch_bounds__(256) void rmsnorm_k(
    const float* __restrict__ x, const float* __restrict__ w,
    float* __restrict__ u)
{
    __shared__ float red[256];
    const long row = blockIdx.x;
    const int  t   = threadIdx.x;
    float v0 = x[row * DMODEL + t];
    float v1 = x[row * DMODEL + 256 + t];
    red[t] = v0 * v0 + v1 * v1;
    __syncthreads();
    for (int s = 128; s > 0; s >>= 1) {
        if (t < s) red[t] += red[t + s];
        __syncthreads();
    }
    float scale = rsqrtf(red[0] * (1.f / DMODEL) + 1e-5f);
    u[row * DMODEL + t]       = v0 * scale * w[t];
    u[row * DMODEL + 256 + t] = v1 * scale * w[256 + t];
}

// causal depthwise conv over L + SiLU; xi lives in uin columns [0,512)
__global__ __launch_bounds__(256) void conv_silu_k(
    const float* __restrict__ uin, const float* __restrict__ w,
    const float* __restrict__ bias, float* __restrict__ xic)
{
    long idx = (long)blockIdx.x * 256 + threadIdx.x;   // (b*L + l)*DM + d
    int  d   = (int)(idx & (DMODEL - 1));
    long bl  = idx >> 9;
    int  l   = (int)(bl & (LSEQ - 1));
    float acc = bias[d];
#pragma unroll
    for (int k = 0; k < KCONV; ++k) {
        int ls = l - (KCONV - 1) + k;
        if (ls >= 0)
            acc += uin[(bl - (KCONV - 1) + k) * (2 * DMODEL) + d] * w[d * KCONV + k];
    }
    xic[idx] = acc / (1.f + expf(-acc));
}

__global__ __launch_bounds__(256) void delta_k(
    float* __restrict__ dl, const float* __restrict__ dtb,
    float* __restrict__ outDelta)
{
    long idx = (long)blockIdx.x * 256 + threadIdx.x;
    int  d   = (int)(idx & (DMODEL - 1));
    float v  = dl[idx] + dtb[d];
    float sp = v > 20.f ? v : log1pf(expf(v));
    dl[idx] = sp;
    if (outDelta) outDelta[idx] = sp;
}

// sequential SSM scan: thread <-> (b,d); 16-state recurrence in registers.
// B/C staged through LDS in chunks of 8 timesteps (256 threads load exactly
// 8x32 floats), cutting barrier count 8x on the serial critical path.
__global__ __launch_bounds__(256) void scan_k(
    const float* __restrict__ delta, const float* __restrict__ xic,
    const float* __restrict__ dbc, const float* __restrict__ A_log,
    const float* __restrict__ Dp, float* __restrict__ y)
{
    __shared__ float sBC[TCHUNK][2 * NSTATE];
    const int b = blockIdx.y;
    const int d = blockIdx.x * 256 + threadIdx.x;
    float An[NSTATE], h[NSTATE];
#pragma unroll
    for (int n = 0; n < NSTATE; ++n) {
        An[n] = -expf(A_log[d * NSTATE + n]);
        h[n]  = 0.f;
    }
    const float dpd = Dp[d];
    const int tc = threadIdx.x >> 5;     // 0..7  (timestep within chunk)
    const int j  = threadIdx.x & 31;     // 0..31 (B/C element)
    for (int t0 = 0; t0 < LSEQ; t0 += TCHUNK) {
        sBC[tc][j] = dbc[((long)b * LSEQ + t0 + tc) * (RKDIM + 2 * NSTATE) + RKDIM + j];
        __syncthreads();
        for (int tt = 0; tt < TCHUNK; ++tt) {
            long row = (long)b * LSEQ + t0 + tt;
            float dlt = delta[row * DMODEL + d];
            float xv  = xic[row * DMODEL + d];
            float dx  = dlt * xv;
            float acc = 0.f;
#pragma unroll
            for (int n = 0; n < NSTATE; ++n) {
                h[n] = expf(dlt * An[n]) * h[n] + dx * sBC[tt][n];
                acc += h[n] * sBC[tt][NSTATE + n];
            }
            y[row * DMODEL + d] = acc + dpd * xv;
        }
        __syncthreads();
    }
}

// g = y * silu(z), z = uin columns [512,1024)
__global__ __launch_bounds__(256) void gate_k(
    const float* __restrict__ y, const float* __restrict__ uin,
    float* __restrict__ g)
{
    long idx = (long)blockIdx.x * 256 + threadIdx.x;
    int  d   = (int)(idx & (DMODEL - 1));
    long row = idx >> 9;
    float z  = uin[row * (2 * DMODEL) + DMODEL + d];
    g[idx]   = y[idx] * (z / (1.f + expf(-z)));
}

__global__ __launch_bounds__(256) void transpose_k(
    const float* __restrict__ in, float* __restrict__ out)
{
    __shared__ float tile[32][33];
    const long base = (long)blockIdx.z * LSEQ * LSEQ;
    int x0 = blockIdx.x * 32, y0 = blockIdx.y * 32;
    for (int r = threadIdx.y; r < 32; r += 8)
        tile[r][threadIdx.x] = in[base + (long)(y0 + r) * LSEQ + x0 + threadIdx.x];
    __syncthreads();
    for (int r = threadIdx.y; r < 32; r += 8)
        out[base + (long)(x0 + r) * LSEQ + y0 + threadIdx.x] = tile[threadIdx.x][r];
}

__global__ __launch_bounds__(256) void rowmean_k(
    const float* __restrict__ x, float* __restrict__ feat, int off)
{
    __shared__ float red[256];
    long row = blockIdx.x;
    int  t   = threadIdx.x;
    red[t] = x[row * DMODEL + t] + x[row * DMODEL + 256 + t];
    __syncthreads();
    for (int s = 128; s > 0; s >>= 1) {
        if (t < s) red[t] += red[t + s];
        __syncthreads();
    }
    if (t == 0)
        feat[(row >> 9) * 1024 + off + (row & (LSEQ - 1))] = red[0] * (1.f / DMODEL);
}

__global__ __launch_bounds__(256) void packBC_k(
    const float* __restrict__ dbc, float* __restrict__ outB,
    float* __restrict__ outC)
{
    long idx = (long)blockIdx.x * 256 + threadIdx.x;   // (b*L + t)*NS + n
    int  n   = (int)(idx & (NSTATE - 1));
    long row = idx >> 4;
    outB[idx] = dbc[row * (RKDIM + 2 * NSTATE) + RKDIM + n];
    outC[idx] = dbc[row * (RKDIM + 2 * NSTATE) + RKDIM + NSTATE + n];
}

__global__ __launch_bounds__(256) void aout_k(
    const float* __restrict__ A_log, float* __restrict__ outA)
{
    int idx = blockIdx.x * 256 + threadIdx.x;          // DMODEL*NSTATE
    outA[idx] = -expf(A_log[idx]);
}

__global__ __launch_bounds__(512) void head_k(
    const float* __restrict__ feat, const float* __restrict__ fc1w,
    const float* __restrict__ fc1b, const float* __restrict__ bng,
    const float* __restrict__ bnb, const float* __restrict__ bnm,
    const float* __restrict__ bnv, const float* __restrict__ headw,
    const float* __restrict__ headb, float* __restrict__ pred)
{
    __shared__ float red[512];
    int b = blockIdx.x, j = threadIdx.x;
    float s = fc1b[j];
    const float* f = feat + (long)b * 1024;
    const float* w = fc1w + (long)j * 1024;
    for (int i = 0; i < 1024; ++i) s += f[i] * w[i];
    float tn = (s - bnm[j]) * rsqrtf(bnv[j] + 1e-5f) * bng[j] + bnb[j];
    float hs = tn / (1.f + expf(-tn));
    red[j] = hs * headw[j];
    __syncthreads();
    for (int st = 256; st > 0; st >>= 1) {
        if (j < st) red[j] += red[j + st];
        __syncthreads();
    }
    if (j == 0) pred[b] = 1.f / (1.f + expf(-(red[0] + headb[0])));
}

// ---------------------------------------------------------------------------
struct MambaParams {
    const float *in_w, *conv_w, *conv_b, *xproj_w, *dt_w, *dt_b;
    const float *A_log, *Dp, *out_w, *norm_w;
};

struct Scratch {
    float *u, *uin, *xic, *dbc, *dl, *y;
};

static void run_mamba(float* x, const MambaParams& P, const Scratch& S,
                      float* outDelta, float* outA, float* outBm, float* outCm,
                      hipStream_t s)
{
    const dim3 blk(32, 8);
    for (int i = 0; i < NLAYER; ++i) {
        const float* in_w    = P.in_w    + (long)i * 2 * DMODEL * DMODEL;
        const float* conv_w  = P.conv_w  + (long)i * DMODEL * KCONV;
        const float* conv_b  = P.conv_b  + (long)i * DMODEL;
        const float* xproj_w = P.xproj_w + (long)i * (RKDIM + 2 * NSTATE) * DMODEL;
        const float* dt_w    = P.dt_w    + (long)i * DMODEL * RKDIM;
        const float* dt_b    = P.dt_b    + (long)i * DMODEL;
        const float* A_log   = P.A_log   + (long)i * DMODEL * NSTATE;
        const float* Dp      = P.Dp      + (long)i * DMODEL;
        const float* out_w   = P.out_w   + (long)i * DMODEL * DMODEL;
        const float* norm_w  = P.norm_w  + (long)i * DMODEL;
        const bool last = (i == NLAYER - 1);

        rmsnorm_k<<<ROWS, 256, 0, s>>>(x, norm_w, S.u);
        // uin = u @ in_w^T : (8192,512)x(512,1024)
        gemm_wmma_f32<true, false><<<dim3(2 * DMODEL / 32, ROWS / 256, 1), blk, 0, s>>>(
            S.u, in_w, S.uin, DMODEL, DMODEL, DMODEL, 2 * DMODEL, 0, 0, 0);
        conv_silu_k<<<(long)ROWS * DMODEL / 256, 256, 0, s>>>(S.uin, conv_w, conv_b, S.xic);
        // dbc = xic @ xproj_w^T : (8192,512)x(512,64)
        gemm_wmma_f32<true, false><<<dim3((RKDIM + 2 * NSTATE) / 32, ROWS / 256, 1), blk, 0, s>>>(
            S.xic, xproj_w, S.dbc, DMODEL, DMODEL, DMODEL, RKDIM + 2 * NSTATE, 0, 0, 0);
        // dl = dt @ dt_w^T : (8192,32 stride 64)x(32,512)
        gemm_wmma_f32<true, false><<<dim3(DMODEL / 32, ROWS / 256, 1), blk, 0, s>>>(
            S.dbc, dt_w, S.dl, RKDIM, RKDIM + 2 * NSTATE, RKDIM, DMODEL, 0, 0, 0);
        delta_k<<<(long)ROWS * DMODEL / 256, 256, 0, s>>>(S.dl, dt_b, last ? outDelta : nullptr);
        scan_k<<<dim3(DMODEL / 256, BATCHN), 256, 0, s>>>(S.dl, S.xic, S.dbc, A_log, Dp, S.y);
        gate_k<<<(long)ROWS * DMODEL / 256, 256, 0, s>>>(S.y, S.uin, S.u);
        // x += g @ out_w^T
        gemm_wmma_f32<true, true><<<dim3(DMODEL / 32, ROWS / 256, 1), blk, 0, s>>>(
            S.u, out_w, x, DMODEL, DMODEL, DMODEL, DMODEL, 0, 0, 0);
        if (last) {
            packBC_k<<<(long)BATCHN * LSEQ * NSTATE / 256, 256, 0, s>>>(S.dbc, outBm, outCm);
            aout_k<<<DMODEL * NSTATE / 256, 256, 0, s>>>(A_log, outA);
        }
    }
}

// ---------------------------------------------------------------------------
extern "C" void kernel_launch(void* const* d_in, const int* in_sizes, int n_in,
                              void* d_out, int out_size, void* d_ws, size_t ws_size,
                              hipStream_t stream)
{
    (void)in_sizes; (void)n_in; (void)out_size; (void)ws_size;
    const float* x_Ab = (const float*)d_in[0];
    const float* x_Ag = (const float*)d_in[1];
    const float* W    = (const float*)d_in[2];
    MambaParams hp = {
        (const float*)d_in[3],  (const float*)d_in[4],  (const float*)d_in[5],
        (const float*)d_in[6],  (const float*)d_in[7],  (const float*)d_in[8],
        (const float*)d_in[9],  (const float*)d_in[10], (const float*)d_in[11],
        (const float*)d_in[12]};
    MambaParams vp = {
        (const float*)d_in[13], (const float*)d_in[14], (const float*)d_in[15],
        (const float*)d_in[16], (const float*)d_in[17], (const float*)d_in[18],
        (const float*)d_in[19], (const float*)d_in[20], (const float*)d_in[21],
        (const float*)d_in[22]};
    const float* fc1_w = (const float*)d_in[23];
    const float* fc1_b = (const float*)d_in[24];
    const float* bn_g  = (const float*)d_in[25];
    const float* bn_b  = (const float*)d_in[26];
    const float* bn_m  = (const float*)d_in[27];
    const float* bn_v  = (const float*)d_in[28];
    const float* hd_w  = (const float*)d_in[29];
    const float* hd_b  = (const float*)d_in[30];

    float* out = (float*)d_out;
    // output layout: pred | h.delta | h.A | h.Bm | h.Cm | v.delta | v.A | v.Bm | v.Cm
    const long D3 = (long)BATCHN * LSEQ * DMODEL;          // 4194304
    const long BC = (long)BATCHN * LSEQ * NSTATE;          // 131072
    float* o_pred = out;
    float* o_hd = out + 16;
    float* o_hA = o_hd + D3;
    float* o_hB = o_hA + DMODEL * NSTATE;
    float* o_hC = o_hB + BC;
    float* o_vd = o_hC + BC;
    float* o_vA = o_vd + D3;
    float* o_vB = o_vA + DMODEL * NSTATE;
    float* o_vC = o_vB + BC;

    float* ws = (float*)d_ws;
    float* Tbuf = ws;                                      // 8192*256
    float* xh   = Tbuf + (long)ROWS * FDIM;                // contacts / horizontal x
    float* xv   = xh + D3;                                 // contacts^T / vertical x
    Scratch S;
    S.u   = xv + D3;
    S.uin = S.u + D3;
    S.xic = S.uin + 2 * D3;
    S.dbc = S.xic + D3;
    S.dl  = S.dbc + (long)ROWS * (RKDIM + 2 * NSTATE);
    S.y   = S.dl + D3;
    float* feat = S.y + D3;                                // 16*1024

    const dim3 blk(32, 8);
    // T = x_Ab @ W  : (8192,256)x(256,256)
    gemm_wmma_f32<false, false><<<dim3(FDIM / 32, ROWS / 256, 1), blk, 0, stream>>>(
        x_Ab, W, Tbuf, FDIM, FDIM, FDIM, FDIM, 0, 0, 0);
    // contacts[b] = T[b] @ x_Ag[b]^T : (512,256)x(256,512), batched
    gemm_wmma_f32<true, false><<<dim3(LSEQ / 32, LSEQ / 256, BATCHN), blk, 0, stream>>>(
        Tbuf, x_Ag, xh, FDIM, FDIM, FDIM, LSEQ,
        (long)LSEQ * FDIM, (long)LSEQ * FDIM, (long)LSEQ * LSEQ);
    // contacts^T for vertical direction (before horizontal mamba mutates xh)
    transpose_k<<<dim3(16, 16, BATCHN), dim3(32, 8), 0, stream>>>(xh, xv);

    run_mamba(xh, hp, S, o_hd, o_hA, o_hB, o_hC, stream);
    rowmean_k<<<ROWS, 256, 0, stream>>>(xh, feat, 0);
    run_mamba(xv, vp, S, o_vd, o_vA, o_vB, o_vC, stream);
    rowmean_k<<<ROWS, 256, 0, stream>>>(xv, feat, LSEQ);

    head_k<<<BATCHN, 512, 0, stream>>>(feat, fc1_w, fc1_b, bn_g, bn_b, bn_m,
                                       bn_v, hd_w, hd_b, o_pred);
}